// CollaborativeLightGCN_80848464380031
// MI455X (gfx1250) — compile-verified
//
#include <hip/hip_runtime.h>
#include <stdint.h>

// ---------------- problem constants (from reference) ----------------
#define NUM_USERS  100000
#define NUM_ITEMS  50000
#define N_NODES    (NUM_USERS + NUM_ITEMS)   // 150000
#define EMBED_DIM  64
#define NUM_LAYERS 3
#define N_EDGES    1000000

#define WAVE        32          // gfx1250 is wave32
#define CONV_BLOCK  256         // 8 waves / block
#define CH          8           // async gather chunk (rows per LDS buffer)

static __device__ __forceinline__ float2 ldg2(const float* p) {
    return *reinterpret_cast<const float2*>(p);
}

// ---------------- 0) zero int buffer ----------------
__global__ void k_zero_i32(int* __restrict__ p, int n) {
    int stride = gridDim.x * blockDim.x;
    for (int i = blockIdx.x * blockDim.x + threadIdx.x; i < n; i += stride)
        p[i] = 0;
}

// ---------------- 1) dst-degree histogram ----------------
__global__ void k_degree(const int* __restrict__ dst, int* __restrict__ cnt) {
    int stride = gridDim.x * blockDim.x;
    for (int e = blockIdx.x * blockDim.x + threadIdx.x; e < N_EDGES; e += stride)
        atomicAdd(&cnt[dst[e]], 1);
}

// ---------------- 2) dinv = deg>0 ? rsqrt(deg) : 0 ----------------
__global__ void k_dinv(const int* __restrict__ cnt, float* __restrict__ dinv) {
    int stride = gridDim.x * blockDim.x;
    for (int i = blockIdx.x * blockDim.x + threadIdx.x; i < N_NODES; i += stride) {
        int c = cnt[i];
        dinv[i] = (c > 0) ? rsqrtf((float)c) : 0.0f;
    }
}

// ---------------- 3) single-block exclusive scan -> rowptr, cursor ----------------
__global__ __launch_bounds__(1024)
void k_scan(const int* __restrict__ cnt, int* __restrict__ rowptr,
            int* __restrict__ cursor) {
    __shared__ int part[1024];
    const int t = threadIdx.x;
    const int chunk = (N_NODES + 1023) / 1024;            // 147
    const int b0 = min(t * chunk, N_NODES);
    const int b1 = min(b0 + chunk, N_NODES);

    int s = 0;
    for (int i = b0; i < b1; ++i) s += cnt[i];
    part[t] = s;
    __syncthreads();

    for (int off = 1; off < 1024; off <<= 1) {
        int v = (t >= off) ? part[t - off] : 0;
        __syncthreads();
        part[t] += v;
        __syncthreads();
    }

    int run = (t > 0) ? part[t - 1] : 0;                  // exclusive base
    for (int i = b0; i < b1; ++i) {
        rowptr[i] = run;
        cursor[i] = run;
        run += cnt[i];
    }
    if (t == 1023) rowptr[N_NODES] = run;                 // == N_EDGES
}

// ---------------- 4) CSR fill: col + per-edge norm weight ----------------
__global__ void k_fill(const int* __restrict__ src, const int* __restrict__ dst,
                       const float* __restrict__ dinv, int* __restrict__ cursor,
                       int* __restrict__ col, float* __restrict__ wn) {
    int stride = gridDim.x * blockDim.x;
    for (int e = blockIdx.x * blockDim.x + threadIdx.x; e < N_EDGES; e += stride) {
        int s = src[e];
        int d = dst[e];
        int p = atomicAdd(&cursor[d], 1);
        col[p] = s;
        wn[p]  = dinv[s] * dinv[d];
    }
}

// ---------------- 5) init: x0 = emb, acc(d_out) = emb ----------------
__global__ void k_init(const float* __restrict__ ew, float* __restrict__ xa,
                       float* __restrict__ acc) {
    const int n4 = (N_NODES * EMBED_DIM) / 4;             // 2.4M float4
    const float4* s4 = reinterpret_cast<const float4*>(ew);
    float4* a4 = reinterpret_cast<float4*>(xa);
    float4* o4 = reinterpret_cast<float4*>(acc);
    int stride = gridDim.x * blockDim.x;
    for (int i = blockIdx.x * blockDim.x + threadIdx.x; i < n4; i += stride) {
        float4 v = s4[i];
        a4[i] = v;
        o4[i] = v;
    }
}

// ---------------- 6) LightGCN layer: async-to-LDS gather, atomic-free ----------------
// One wave per destination node; each lane owns dims [2*lane, 2*lane+1].
// Source rows are gathered via gfx1250 ASYNCcnt global->LDS DMA into a
// wave-private double buffer; chunk c's loads overlap chunk c-1's FMAs.
// VDST LDS address = low 32 bits of the flat LDS pointer (ISA 10.2 aperture map).
__global__ __launch_bounds__(CONV_BLOCK)
void k_conv(const float* __restrict__ x, float* __restrict__ y,
            float* __restrict__ acc, const int* __restrict__ rowptr,
            const int* __restrict__ col, const float* __restrict__ wn) {
    __shared__ float smem[(CONV_BLOCK / WAVE) * 2 * CH * EMBED_DIM];   // 32 KB
    const int node = (int)((blockIdx.x * (unsigned)CONV_BLOCK + threadIdx.x) >> 5);
    const int lane = threadIdx.x & (WAVE - 1);
    if (node >= N_NODES) return;                  // whole wave exits together

    float* wavebuf = &smem[(threadIdx.x >> 5) * (2 * CH * EMBED_DIM)];
    const int lane_off = lane * 2;

    const int beg = rowptr[node];
    const int end = rowptr[node + 1];
    const int deg = end - beg;
    const int nfull = deg >> 3;                   // full chunks of CH=8
    const int rem   = deg & (CH - 1);

    float ax = 0.0f, ay = 0.0f;

    auto issue = [&](int c, int m) {
        const int jb = beg + c * CH;
        __builtin_prefetch(&col[jb + CH], 0, 0);  // global_prefetch_b8
        __builtin_prefetch(&wn[jb + CH], 0, 0);
        float* db = wavebuf + (c & 1) * (CH * EMBED_DIM);
        for (int k = 0; k < m; ++k) {
            const int s = col[jb + k];
            const float* gp = x + (size_t)s * EMBED_DIM + lane_off;
            uint32_t lds = (uint32_t)(uintptr_t)(void*)(db + k * EMBED_DIM + lane_off);
            asm volatile("global_load_async_to_lds_b64 %0, %1, off"
                         :: "v"(lds), "v"(gp) : "memory");
        }
    };
    auto consume = [&](int c, int m) {
        const int jb = beg + c * CH;
        const float* sb = wavebuf + (c & 1) * (CH * EMBED_DIM);
        for (int k = 0; k < m; ++k) {
            const float wv = wn[jb + k];
            const float2 r = *reinterpret_cast<const float2*>(sb + k * EMBED_DIM + lane_off);
            ax = fmaf(r.x, wv, ax);
            ay = fmaf(r.y, wv, ay);
        }
    };

    if (nfull > 0) {
        issue(0, CH);
        for (int c = 1; c < nfull; ++c) {
            issue(c, CH);
            // <= CH outstanding  =>  chunk c-1 fully landed (async loads complete in order)
            asm volatile("s_wait_asynccnt 0x8" ::: "memory");
            consume(c - 1, CH);
        }
        if (rem > 0) {
            issue(nfull, rem);                    // tail overlaps last full chunk's FMAs
            switch (rem) {                        // wait <= rem  =>  last full chunk landed
                case 1: asm volatile("s_wait_asynccnt 0x1" ::: "memory"); break;
                case 2: asm volatile("s_wait_asynccnt 0x2" ::: "memory"); break;
                case 3: asm volatile("s_wait_asynccnt 0x3" ::: "memory"); break;
                case 4: asm volatile("s_wait_asynccnt 0x4" ::: "memory"); break;
                case 5: asm volatile("s_wait_asynccnt 0x5" ::: "memory"); break;
                case 6: asm volatile("s_wait_asynccnt 0x6" ::: "memory"); break;
                default: asm volatile("s_wait_asynccnt 0x7" ::: "memory"); break;
            }
            consume(nfull - 1, CH);
            asm volatile("s_wait_asynccnt 0x0" ::: "memory");
            consume(nfull, rem);
        } else {
            asm volatile("s_wait_asynccnt 0x0" ::: "memory");
            consume(nfull - 1, CH);
        }
    } else if (rem > 0) {
        issue(0, rem);                            // typical node: deg<8, issue-all-then-wait
        asm volatile("s_wait_asynccnt 0x0" ::: "memory");
        consume(0, rem);
    }

    const size_t o = (size_t)node * EMBED_DIM + lane_off;
    float2 outv; outv.x = ax; outv.y = ay;
    *reinterpret_cast<float2*>(y + o) = outv;     // next layer input
    float2 c2 = ldg2(acc + o);                    // fused acc += emb
    c2.x += ax; c2.y += ay;
    *reinterpret_cast<float2*>(acc + o) = c2;
}

// ---------------- 7) final scale by 1/(NUM_LAYERS+1) ----------------
__global__ void k_scale(float* __restrict__ out) {
    const int n4 = (N_NODES * EMBED_DIM) / 4;
    float4* o4 = reinterpret_cast<float4*>(out);
    int stride = gridDim.x * blockDim.x;
    for (int i = blockIdx.x * blockDim.x + threadIdx.x; i < n4; i += stride) {
        float4 v = o4[i];
        v.x *= 0.25f; v.y *= 0.25f; v.z *= 0.25f; v.w *= 0.25f;
        o4[i] = v;
    }
}

// ---------------- host side ----------------
extern "C" void kernel_launch(void* const* d_in, const int* in_sizes, int n_in,
                              void* d_out, int out_size, void* d_ws, size_t ws_size,
                              hipStream_t stream) {
    (void)in_sizes; (void)n_in; (void)out_size; (void)ws_size;

    const int*   edge = (const int*)d_in[0];      // (2, N_EDGES) row-major int32
    const int*   src  = edge;
    const int*   dst  = edge + N_EDGES;
    const float* ew   = (const float*)d_in[1];    // (N_NODES, 64) f32
    float*       out  = (float*)d_out;            // (N_NODES, 64) f32 (users then items)

    // workspace carve-out (~88 MB total)
    char* ws = (char*)d_ws;
    size_t off = 0;
    auto carve = [&](size_t bytes) -> void* {
        void* p = ws + off;
        off += (bytes + 255) & ~(size_t)255;
        return p;
    };
    int*   cnt    = (int*)  carve((size_t)N_NODES * 4);
    float* dinv   = (float*)carve((size_t)N_NODES * 4);
    int*   rowptr = (int*)  carve((size_t)(N_NODES + 1) * 4);
    int*   cursor = (int*)  carve((size_t)N_NODES * 4);
    int*   col    = (int*)  carve((size_t)N_EDGES * 4);
    float* wn     = (float*)carve((size_t)N_EDGES * 4);
    float* xa     = (float*)carve((size_t)N_NODES * EMBED_DIM * 4);
    float* xb     = (float*)carve((size_t)N_NODES * EMBED_DIM * 4);

    k_zero_i32<<<512, 256, 0, stream>>>(cnt, N_NODES);
    k_degree  <<<2048, 256, 0, stream>>>(dst, cnt);
    k_dinv    <<<512, 256, 0, stream>>>(cnt, dinv);
    k_scan    <<<1, 1024, 0, stream>>>(cnt, rowptr, cursor);
    k_fill    <<<2048, 256, 0, stream>>>(src, dst, dinv, cursor, col, wn);
    k_init    <<<2048, 256, 0, stream>>>(ew, xa, out);

    float* bufs[2] = {xa, xb};
    const int conv_grid = (N_NODES * WAVE + CONV_BLOCK - 1) / CONV_BLOCK; // 18750
    for (int l = 0; l < NUM_LAYERS; ++l) {
        k_conv<<<conv_grid, CONV_BLOCK, 0, stream>>>(
            bufs[l & 1], bufs[(l + 1) & 1], out, rowptr, col, wn);
    }
    k_scale<<<2048, 256, 0, stream>>>(out);
}